// TritonFFT_53584011985647
// MI455X (gfx1250) — compile-verified
//
#include <hip/hip_runtime.h>

typedef float v2f __attribute__((ext_vector_type(2)));
typedef float v8f __attribute__((ext_vector_type(8)));

// cos(-2*pi*m/8) for m in 0..7 : {1, r, 0, -r, -1, -r, 0, r}
__device__ __forceinline__ float wre_tab(int m) {
    const float r = 0.70710678118654752440f;
    float mag = (m & 1) ? r : (((m & 3) == 0) ? 1.0f : 0.0f);
    bool neg = (m >= 3) && (m <= 5);
    return neg ? -mag : mag;
}
// sin(-2*pi*m/8) for m in 0..7 : {0, -r, -1, -r, 0, r, 1, r}
__device__ __forceinline__ float wim_tab(int m) {
    const float r = 0.70710678118654752440f;
    float mag = (m & 1) ? r : (((m & 3) == 2) ? 1.0f : 0.0f);
    bool neg = (m >= 1) && (m <= 3);
    return neg ? -mag : mag;
}

// Combined 16x16 block matrix: [[Wre, Wim], [-Wim, Wre]]
// k indexes [x_re(0..7) | x_im(0..7)], n indexes [y_re(0..7) | y_im(0..7)]
__device__ __forceinline__ float bmat(int k, int n) {
    int m = ((k & 7) * (n & 7)) & 7;
    float re = wre_tab(m);
    float im = wim_tab(m);
    if (n < 8) return (k < 8) ? re : -im;
    else       return (k < 8) ? im : re;
}

__global__ void fft8_wmma_kernel(const float* __restrict__ xre,
                                 const float* __restrict__ xim,
                                 float* __restrict__ yre,
                                 float* __restrict__ yim,
                                 int numTiles) {
    const int lane          = threadIdx.x & 31;
    const int waveInBlock   = threadIdx.x >> 5;
    const int wavesPerBlock = blockDim.x >> 5;
    const int gWave         = blockIdx.x * wavesPerBlock + waveInBlock;
    const int nWaves        = gridDim.x * wavesPerBlock;

    const int half = lane >> 4;   // 0: K pair {4c,4c+1}, 1: K pair {4c+2,4c+3}
    const int l15  = lane & 15;

    // Build the wave-invariant B matrix fragments once.
    // B chunk c, VGPR v, lane L: K = 4c + 2*half + v, N = L&15
    v2f B0, B1, B2, B3;
    {
        int k0 = 0 + 2 * half;
        int k1 = 4 + 2 * half;
        int k2 = 8 + 2 * half;
        int k3 = 12 + 2 * half;
        B0.x = bmat(k0 + 0, l15); B0.y = bmat(k0 + 1, l15);
        B1.x = bmat(k1 + 0, l15); B1.y = bmat(k1 + 1, l15);
        B2.x = bmat(k2 + 0, l15); B2.y = bmat(k2 + 1, l15);
        B3.x = bmat(k3 + 0, l15); B3.y = bmat(k3 + 1, l15);
    }

    for (int tile = gWave; tile < numTiles; tile += nWaves) {
        const long base = (long)tile * 16;
        const long row  = base + l15;
        const float* pre = xre + row * 8 + half * 2;
        const float* pim = xim + row * 8 + half * 2;

        // A fragments: chunk c covers K = 4c..4c+3; this lane holds 2 of them.
        v2f a0 = *(const v2f*)(pre + 0);   // K 0..3  -> x_re cols {0,1}/{2,3}
        v2f a1 = *(const v2f*)(pre + 4);   // K 4..7  -> x_re cols {4,5}/{6,7}
        v2f a2 = *(const v2f*)(pim + 0);   // K 8..11 -> x_im cols {0,1}/{2,3}
        v2f a3 = *(const v2f*)(pim + 4);   // K 12..15-> x_im cols {4,5}/{6,7}

        v8f acc = {};
        acc = __builtin_amdgcn_wmma_f32_16x16x4_f32(false, a0, false, B0,
                                                    (short)0, acc, false, false);
        acc = __builtin_amdgcn_wmma_f32_16x16x4_f32(false, a1, false, B1,
                                                    (short)0, acc, false, false);
        acc = __builtin_amdgcn_wmma_f32_16x16x4_f32(false, a2, false, B2,
                                                    (short)0, acc, false, false);
        acc = __builtin_amdgcn_wmma_f32_16x16x4_f32(false, a3, false, B3,
                                                    (short)0, acc, false, false);

        // D layout: VGPR v -> M = v + 8*half, lane -> N = l15.
        // N<8 -> y_re column N ; N>=8 -> y_im column N-8.
        const int col = l15 & 7;
        float* outCol = (l15 < 8) ? yre : yim;
        #pragma unroll
        for (int v = 0; v < 8; ++v) {
            long m = base + v + half * 8;
            __builtin_nontemporal_store(acc[v], outCol + m * 8 + col);
        }
    }
}

// Scalar tail for row counts not divisible by 16 (not hit for 2,000,000 rows).
__global__ void fft8_tail_kernel(const float* __restrict__ xre,
                                 const float* __restrict__ xim,
                                 float* __restrict__ yre,
                                 float* __restrict__ yim,
                                 long startRow, long rows) {
    long r = startRow + (long)blockIdx.x * blockDim.x + threadIdx.x;
    if (r >= rows) return;
    float re[8], im[8];
    #pragma unroll
    for (int n = 0; n < 8; ++n) { re[n] = xre[r * 8 + n]; im[n] = xim[r * 8 + n]; }
    #pragma unroll
    for (int k = 0; k < 8; ++k) {
        float sre = 0.0f, sim = 0.0f;
        #pragma unroll
        for (int n = 0; n < 8; ++n) {
            int m = (k * n) & 7;
            float wr = wre_tab(m), wi = wim_tab(m);
            sre += re[n] * wr - im[n] * wi;
            sim += re[n] * wi + im[n] * wr;
        }
        yre[r * 8 + k] = sre;
        yim[r * 8 + k] = sim;
    }
}

extern "C" void kernel_launch(void* const* d_in, const int* in_sizes, int n_in,
                              void* d_out, int out_size, void* d_ws, size_t ws_size,
                              hipStream_t stream) {
    const float* xre = (const float*)d_in[0];
    const float* xim = (const float*)d_in[1];
    const long rows = (long)in_sizes[0] / 8;
    float* yre = (float*)d_out;
    float* yim = (float*)d_out + rows * 8;

    const long tiles = rows / 16;
    if (tiles > 0) {
        const int  block         = 256;           // 8 waves
        const int  wavesPerBlock = block / 32;
        long targetWaves = (tiles + 7) / 8;       // ~8 tiles per wave
        long blocks = (targetWaves + wavesPerBlock - 1) / wavesPerBlock;
        if (blocks < 1) blocks = 1;
        if (blocks > 65535) blocks = 65535;       // grid-stride covers the rest
        fft8_wmma_kernel<<<(int)blocks, block, 0, stream>>>(
            xre, xim, yre, yim, (int)tiles);
    }
    const long rem = rows - tiles * 16;
    if (rem > 0) {
        const int tb = 256;
        long tg = (rem + tb - 1) / tb;
        fft8_tail_kernel<<<(int)tg, tb, 0, stream>>>(
            xre, xim, yre, yim, tiles * 16, rows);
    }
}